// CrossEntropyLoss_2585570312585
// MI455X (gfx1250) — compile-verified
//
#include <hip/hip_runtime.h>

// Problem constants (fixed shapes from the reference: [1,5,256,384]).
#define NCLS   5
#define HW     98304          // 256*384 pixels
#define NBLK   96             // 96 blocks * 256 threads * 4 pixels = 98304
#define NTHR   256
#define EPSF   1e-8f

typedef __attribute__((ext_vector_type(2))) float v2f;
typedef __attribute__((ext_vector_type(8))) float v8f;

// Stage 1: per-pixel loss, per-thread accumulation (4 pixels via float4 loads),
// block reduction of the 256 partials done with V_WMMA_F32_16X16X4_F32
// (ones-matmul reduction), one partial per block written to d_ws.
__global__ void __launch_bounds__(NTHR)
ce_loss_main(const float* __restrict__ pred,
             const float* __restrict__ gold,
             const float* __restrict__ weight,
             const float* __restrict__ cwl,
             float* __restrict__ partials) {
    __shared__ float sdata[NTHR];

    const int tid = threadIdx.x;
    const int t   = blockIdx.x * NTHR + tid;   // 0 .. 24575
    const int j   = t * 4;                     // base pixel, 16B aligned

    // Class weights with the torch-loop substitution: cw[i]==0 -> cw[0].
    float cw[NCLS];
#pragma unroll
    for (int i = 0; i < NCLS; ++i) cw[i] = cwl[i];
#pragma unroll
    for (int i = 1; i < NCLS; ++i) cw[i] = (cw[i] == 0.0f) ? cw[0] : cw[i];

    // Coalesced b128 loads: each byte of pred/gold/weight touched exactly once.
    float p_[NCLS][4], g_[NCLS][4];
#pragma unroll
    for (int c = 0; c < NCLS; ++c) {
        float4 pv = *(const float4*)(pred + (size_t)c * HW + j);
        float4 gv = *(const float4*)(gold + (size_t)c * HW + j);
        p_[c][0] = pv.x; p_[c][1] = pv.y; p_[c][2] = pv.z; p_[c][3] = pv.w;
        g_[c][0] = gv.x; g_[c][1] = gv.y; g_[c][2] = gv.z; g_[c][3] = gv.w;
    }
    float4 wv = *(const float4*)(weight + j);
    float w_[4] = { wv.x, wv.y, wv.z, wv.w };

    float acc = 0.0f;
#pragma unroll
    for (int k = 0; k < 4; ++k) {
        float bp = p_[0][k]; int ap = 0;      // argmax(pred), first-max wins
        float bg = g_[0][k]; int ag = 0;      // argmax(gold)
        float dot = g_[0][k] * __logf(p_[0][k] + EPSF);
#pragma unroll
        for (int c = 1; c < NCLS; ++c) {
            float p = p_[c][k];
            float g = g_[c][k];
            if (p > bp) { bp = p; ap = c; }
            if (g > bg) { bg = g; ag = c; }
            dot += g * __logf(p + EPSF);
        }
        int pfp = (ap == ag) ? 0 : ap;        // pred_fp_fn
        pfp     = (ag > 0)   ? 0 : pfp;       // pred_fp
        float wfp = cw[0];                    // cndmask chain (no scratch array)
        wfp = (pfp == 1) ? cw[1] : wfp;
        wfp = (pfp == 2) ? cw[2] : wfp;
        wfp = (pfp == 3) ? cw[3] : wfp;
        wfp = (pfp == 4) ? cw[4] : wfp;
        acc -= (w_[k] + wfp) * dot;           // loss = -(w + wfp) * dot
    }

    sdata[tid] = acc;
    __syncthreads();

    // Wave 0 (tid<32 is wave-uniform -> EXEC all ones inside) reduces all 256
    // partials with 4 accumulating f32 WMMAs: D = A x ones + C gives row-sums.
    if (tid < 32) {
        const int lane = tid;
        const int m    = lane & 15;           // A row
        const int khi  = (lane >> 4) << 1;    // K base: lanes 0-15 -> K0/K1, 16-31 -> K2/K3
        v2f b; b[0] = 1.0f; b[1] = 1.0f;      // B = ones (4x16)
        v8f cacc = {};                        // C = 0, accumulates across chunks
#pragma unroll
        for (int chunk = 0; chunk < 4; ++chunk) {
            const int base = chunk * 64;
            v2f a;                            // A[m][k] = sdata[base + m*4 + k]
            a[0] = sdata[base + m * 4 + khi];
            a[1] = sdata[base + m * 4 + khi + 1];
            cacc = __builtin_amdgcn_wmma_f32_16x16x4_f32(
                false, a, false, b, (short)0, cacc, false, false);
        }
        // D[m][n] identical for all n: lanes 0-15 hold rows 0..7 in v0..7,
        // lanes 16-31 hold rows 8..15. Sum 8 regs, then fold the two halves.
        float s = 0.0f;
#pragma unroll
        for (int r = 0; r < 8; ++r) s += cacc[r];
        s += __shfl_xor(s, 16, 32);
        if (lane == 0) partials[blockIdx.x] = s;
    }
}

// Stage 2: deterministic serial fold of the 96 block partials + mean.
__global__ void ce_loss_final(const float* __restrict__ partials,
                              float* __restrict__ out) {
    if (threadIdx.x == 0 && blockIdx.x == 0) {
        float s = 0.0f;
        for (int i = 0; i < NBLK; ++i) s += partials[i];
        out[0] = s * (1.0f / (float)HW);
    }
}

extern "C" void kernel_launch(void* const* d_in, const int* in_sizes, int n_in,
                              void* d_out, int out_size, void* d_ws, size_t ws_size,
                              hipStream_t stream) {
    const float* pred   = (const float*)d_in[0];  // [1,5,256,384]
    const float* gold   = (const float*)d_in[1];  // [1,5,256,384]
    const float* weight = (const float*)d_in[2];  // [1,256,384]
    const float* cwl    = (const float*)d_in[3];  // [1,5]
    float* partials     = (float*)d_ws;           // NBLK floats of scratch
    float* out          = (float*)d_out;          // scalar

    ce_loss_main<<<NBLK, NTHR, 0, stream>>>(pred, gold, weight, cwl, partials);
    ce_loss_final<<<1, 32, 0, stream>>>(partials, out);
}